// Conditionals_82995948028288
// MI455X (gfx1250) — compile-verified
//
#include <hip/hip_runtime.h>

// ---------------------------------------------------------------------------
// Causal-network sequential evaluation, restructured for MI455X (gfx1250):
//  - A pre-transposed+compressed to u8 (contiguous step reads, ~4x less HBM)
//  - batches bucketed per (step, node) so ~32 batches share W1[node] in LDS
//  - bucket MLP layer 1 as fully-unrolled WMMA f32 16x16x4 GEMM (wave32),
//    B operand resident in VGPRs, A fragments streamed from LDS
//  - 128 sequential step launches provide the inter-step ordering
// ---------------------------------------------------------------------------

typedef __attribute__((ext_vector_type(2))) float v2f;
typedef __attribute__((ext_vector_type(8))) float v8f;

constexpr int N_NODES = 128;
constexpr int BATCH   = 4096;
constexpr int HIDDEN  = 32;
constexpr int D_IN    = N_NODES + 1;   // 129
constexpr int K_PAD   = 132;           // 33 chunks of 4
constexpr int NCHUNK  = K_PAD / 4;     // 33
constexpr int W1_STR  = 48;            // LDS row stride for W1 tile (bank-friendly)
constexpr int MTILE   = 32;            // batches per block tile (2 WMMA M-subtiles)
constexpr float SLOPE = 0.01f;         // jax.nn.leaky_relu default

// ---------------- init outputs0: u at do_idx, else 0 ----------------
__global__ void init_outputs_kernel(const float* __restrict__ u,
                                    const int* __restrict__ do_idxs,
                                    float* __restrict__ outputs) {
  int i = blockIdx.x * blockDim.x + threadIdx.x;
  if (i >= BATCH * N_NODES) return;
  int b = i / N_NODES, n = i % N_NODES;
  outputs[i] = (do_idxs[b] == n) ? u[b] : 0.0f;
}

// ---------------- bucket building ----------------
__global__ void zero_counts_kernel(int* __restrict__ counts, int* __restrict__ cursors) {
  int i = blockIdx.x * blockDim.x + threadIdx.x;
  if (i < N_NODES * N_NODES) { counts[i] = 0; cursors[i] = 0; }
}

__global__ void count_kernel(const int* __restrict__ order, int* __restrict__ counts) {
  int i = blockIdx.x * blockDim.x + threadIdx.x;   // i = b*128 + t
  if (i >= BATCH * N_NODES) return;
  int t = i % N_NODES;
  int n = order[i];
  atomicAdd(&counts[t * N_NODES + n], 1);
}

__global__ void prefix_kernel(const int* __restrict__ counts, int* __restrict__ offsets) {
  int t = blockIdx.x;
  __shared__ int sc[N_NODES];
  sc[threadIdx.x] = counts[t * N_NODES + threadIdx.x];
  __syncthreads();
  if (threadIdx.x == 0) {
    int s = 0;
    for (int n = 0; n < N_NODES; ++n) { int c = sc[n]; sc[n] = s; s += c; }
  }
  __syncthreads();
  offsets[t * N_NODES + threadIdx.x] = sc[threadIdx.x];
}

__global__ void scatter_kernel(const int* __restrict__ order,
                               const int* __restrict__ offsets,
                               int* __restrict__ cursors,
                               int* __restrict__ bucket) {
  int i = blockIdx.x * blockDim.x + threadIdx.x;   // i = b*128 + t
  if (i >= BATCH * N_NODES) return;
  int b = i / N_NODES, t = i % N_NODES;
  int n = order[i];
  int idx = atomicAdd(&cursors[t * N_NODES + n], 1);
  bucket[t * BATCH + offsets[t * N_NODES + n] + idx] = b;
}

// ---------------- A[b,d,n] -> A8[b,n,d] (transpose + compress to u8) -------
__global__ void transpose_kernel(const float* __restrict__ A, unsigned char* __restrict__ A8) {
  int b  = blockIdx.z;
  int ni = blockIdx.x, di = blockIdx.y;
  __shared__ float tile[32][33];
  int tx = threadIdx.x, ty = threadIdx.y;       // (32, 8)
  const float* Ab = A + (size_t)b * N_NODES * N_NODES;
  unsigned char* A8b = A8 + (size_t)b * N_NODES * N_NODES;
#pragma unroll
  for (int k = 0; k < 4; ++k) {
    int dl = ty + 8 * k;
    tile[dl][tx] = Ab[(size_t)(di * 32 + dl) * N_NODES + ni * 32 + tx];   // coalesced in n
  }
  __syncthreads();
#pragma unroll
  for (int k = 0; k < 4; ++k) {
    int nl = ty + 8 * k;
    A8b[(size_t)(ni * 32 + nl) * N_NODES + di * 32 + tx] =
        (unsigned char)(tile[tx][nl] != 0.0f ? 1 : 0);                     // coalesced in d
  }
}

// ---------------- one sequential step: all 128 node-buckets in parallel ----
__global__ void __launch_bounds__(128)
step_kernel(int t,
            const float* __restrict__ A, const unsigned char* __restrict__ A8, int useA8,
            const float* __restrict__ x,
            const float* __restrict__ W1, const float* __restrict__ b1,
            const float* __restrict__ W2, const float* __restrict__ b2,
            const int* __restrict__ do_idxs,
            const int* __restrict__ counts, const int* __restrict__ offsets,
            const int* __restrict__ bucket,
            float* __restrict__ outputs) {
  int n = blockIdx.x;                       // node for this bucket
  int cnt = counts[t * N_NODES + n];
  if (cnt == 0) return;
  const int* blist = bucket + t * BATCH + offsets[t * N_NODES + n];

  __shared__ float w1s[K_PAD * W1_STR];     // W1[n] as [K][HIDDEN], padded rows zeroed
  __shared__ float ins[MTILE * K_PAD];      // 32-batch tile of masked inputs
  __shared__ float hs[MTILE * 33];          // hidden activations
  __shared__ float w2s[HIDDEN];
  __shared__ float b1s[HIDDEN];
  __shared__ float b2s;
  __shared__ int   bls[MTILE];

  int tid  = threadIdx.x;                   // 0..127 (4 waves)
  int wave = tid >> 5;
  int lane = tid & 31;
  int mt   = wave >> 1;                     // M-subtile 0/1  (rows 16*mt .. +16)
  int nt   = wave & 1;                      // N-subtile 0/1  (hidden 16*nt .. +16)

  const float* W1n = W1 + (size_t)n * D_IN * HIDDEN;
  for (int i = tid; i < K_PAD * HIDDEN; i += 128) {
    int k = i / HIDDEN, j = i % HIDDEN;
    w1s[k * W1_STR + j] = (k < D_IN) ? W1n[k * HIDDEN + j] : 0.0f;
  }
  if (tid < HIDDEN) { w2s[tid] = W2[n * HIDDEN + tid]; b1s[tid] = b1[n * HIDDEN + tid]; }
  if (tid == 0) b2s = b2[n];
  __syncthreads();

  // --- Preload B operand (W1[n]) into registers: invariant across M tiles.
  // B 4x16 f32 layout: lane = N (mod 16), half-waves carry K = {0,1} / {2,3}.
  int ncol = 16 * nt + (lane & 15);
  int koff = (lane >> 4) * 2;
  v2f bf[NCHUNK];
#pragma unroll
  for (int c = 0; c < NCHUNK; ++c) {
    int kb = c * 4 + koff;
    bf[c].x = w1s[kb * W1_STR + ncol];
    bf[c].y = w1s[(kb + 1) * W1_STR + ncol];
  }

  int Mrow  = 16 * mt + (lane & 15);
  int mbase = 16 * mt + (lane >> 4) * 8;
  float bj  = b1s[ncol];

  int ntiles = (cnt + MTILE - 1) / MTILE;
  for (int tile = 0; tile < ntiles; ++tile) {
    int m0 = tile * MTILE;
    int mcnt = min(MTILE, cnt - m0);
    if (tid < MTILE) bls[tid] = (tid < mcnt) ? blist[m0 + tid] : 0;
    __syncthreads();

    // Build ins[m][d] = mask(A[b,d,n]) * outputs[b,d]; ins[m][128] = x[b,n]; pad 0
    for (int i = tid; i < MTILE * K_PAD; i += 128) {
      int m = i / K_PAD, d = i % K_PAD;
      float v = 0.0f;
      if (m < mcnt) {
        int b = bls[m];
        if (d < N_NODES) {
          bool conn;
          if (useA8)
            conn = A8[(size_t)b * N_NODES * N_NODES + (size_t)n * N_NODES + d] != 0;
          else
            conn = A[(size_t)b * N_NODES * N_NODES + (size_t)d * N_NODES + n] != 0.0f;
          v = conn ? outputs[b * N_NODES + d] : 0.0f;
        } else if (d == N_NODES) {
          v = x[b * N_NODES + n];
        }
      }
      ins[i] = v;
    }
    __syncthreads();

    // GEMM (32 x K_PAD) @ (K_PAD x 32) via V_WMMA_F32_16X16X4_F32, 1 subtile/wave.
    if (16 * mt < mcnt) {                    // wave-uniform; skip empty M-subtile
      v8f acc = {0.f, 0.f, 0.f, 0.f, 0.f, 0.f, 0.f, 0.f};
#pragma unroll
      for (int c = 0; c < NCHUNK; ++c) {
        int kb = c * 4 + koff;
        v2f a;
        a.x = ins[Mrow * K_PAD + kb];
        a.y = ins[Mrow * K_PAD + kb + 1];
        acc = __builtin_amdgcn_wmma_f32_16x16x4_f32(
            /*neg_a=*/false, a, /*neg_b=*/false, bf[c],
            /*c_mod=*/(short)0, acc, /*reuse_a=*/false, /*reuse_b=*/false);
      }
      // bias + leaky relu; D layout: VGPR r -> M = mbase + r, N = ncol
#pragma unroll
      for (int r = 0; r < 8; ++r) {
        float v = acc[r] + bj;
        v = (v > 0.0f) ? v : SLOPE * v;
        hs[(mbase + r) * 33 + ncol] = v;
      }
    }
    __syncthreads();

    // output layer + do-intervention, one thread per batch row
    if (tid < mcnt) {
      float s = b2s;
#pragma unroll
      for (int j = 0; j < HIDDEN; ++j) s += hs[tid * 33 + j] * w2s[j];
      int b = bls[tid];
      if (do_idxs[b] != n) outputs[b * N_NODES + n] = s;  // else keep u[b]
    }
    __syncthreads();
  }
}

// ---------------------------------------------------------------------------
extern "C" void kernel_launch(void* const* d_in, const int* in_sizes, int n_in,
                              void* d_out, int out_size, void* d_ws, size_t ws_size,
                              hipStream_t stream) {
  const float* x   = (const float*)d_in[0];
  const float* A   = (const float*)d_in[1];
  const float* u   = (const float*)d_in[2];
  const float* W1  = (const float*)d_in[3];
  const float* b1  = (const float*)d_in[4];
  const float* W2  = (const float*)d_in[5];
  const float* b2  = (const float*)d_in[6];
  const int*   order   = (const int*)d_in[7];
  const int*   do_idxs = (const int*)d_in[8];
  float* outputs = (float*)d_out;

  // workspace carve-up
  char* ws = (char*)d_ws;
  size_t off = 0;
  auto carve = [&](size_t bytes) { char* p = ws + off; off = (off + bytes + 255) & ~(size_t)255; return p; };
  int* counts  = (int*)carve((size_t)N_NODES * N_NODES * 4);
  int* offsets = (int*)carve((size_t)N_NODES * N_NODES * 4);
  int* cursors = (int*)carve((size_t)N_NODES * N_NODES * 4);
  int* bucket  = (int*)carve((size_t)N_NODES * BATCH * 4);
  size_t a8_bytes = (size_t)BATCH * N_NODES * N_NODES;   // 64 MB
  unsigned char* A8 = nullptr;
  int useA8 = (off + a8_bytes <= ws_size) ? 1 : 0;
  if (useA8) A8 = (unsigned char*)carve(a8_bytes);

  // 1) outputs0
  init_outputs_kernel<<<(BATCH * N_NODES + 255) / 256, 256, 0, stream>>>(u, do_idxs, outputs);
  // 2) buckets
  zero_counts_kernel<<<(N_NODES * N_NODES + 255) / 256, 256, 0, stream>>>(counts, cursors);
  count_kernel<<<(BATCH * N_NODES + 255) / 256, 256, 0, stream>>>(order, counts);
  prefix_kernel<<<N_NODES, N_NODES, 0, stream>>>(counts, offsets);
  scatter_kernel<<<(BATCH * N_NODES + 255) / 256, 256, 0, stream>>>(order, offsets, cursors, bucket);
  // 3) transpose+compress A
  if (useA8) {
    dim3 grid(N_NODES / 32, N_NODES / 32, BATCH), blk(32, 8);
    transpose_kernel<<<grid, blk, 0, stream>>>(A, A8);
  }
  // 4) 128 sequential steps, 128 node-buckets each
  for (int t = 0; t < N_NODES; ++t) {
    step_kernel<<<N_NODES, 128, 0, stream>>>(t, A, A8, useA8, x, W1, b1, W2, b2,
                                             do_idxs, counts, offsets, bucket, outputs);
  }
}